// NormRouter_20306605375575
// MI455X (gfx1250) — compile-verified
//
#include <hip/hip_runtime.h>
#include <math.h>

typedef __attribute__((ext_vector_type(2))) float v2f;
typedef __attribute__((ext_vector_type(8))) float v8f;

#define T_TOK  16384
#define DMODEL 2048
#define NEXP   64
#define TOPK   8
#define PFDIST 1024   // prefetch distance in floats (4 KB ahead per row stream)

// ---------------------------------------------------------------------------
// Kernel 1: router projection  logits_clean[t][e] = sum_k h[t][k] * W[e][k]
// Wave tile: 32 rows x 64 experts = 2 M-tiles x 4 N-tiles of 16x16 f32 WMMA.
// K-loop: V_WMMA_F32_16X16X4_F32, K advances by 4 per step.
//
// A fragment (16x4 f32, ISA 7.12.2): lane<16 holds (M=lane, K=k0..k0+1),
// lane>=16 holds (M=lane-16, K=k0+2..k0+3) -> one b64 load per lane.
// B fragment (4x16): lane holds (K=k0+2*(lane>>4)+{0,1}, N=lane&15),
// with B[k][e] = W[e][k] -> one b64 load per lane from row (e) of W.
// C/D (16x16 f32): VGPR g, lanes0-15 -> (M=g, N=lane), lanes16-31 -> (M=g+8).
// ---------------------------------------------------------------------------
__global__ __launch_bounds__(256) void router_gemm_wmma_kernel(
    const float* __restrict__ h,
    const float* __restrict__ W,
    float* __restrict__ logits_clean)
{
    const int lane  = threadIdx.x & 31;
    const int wave  = threadIdx.x >> 5;
    const int l15   = lane & 15;
    const int khalf = (lane >> 4) << 1;          // 0 or 2
    const int row0  = blockIdx.x * 256 + wave * 32;

    const float* pa0 = h + (size_t)(row0 + l15)      * DMODEL + khalf;
    const float* pa1 = h + (size_t)(row0 + 16 + l15) * DMODEL + khalf;
    const float* pb0 = W + (size_t)(l15)             * DMODEL + khalf;
    const float* pb1 = W + (size_t)(16 + l15)        * DMODEL + khalf;
    const float* pb2 = W + (size_t)(32 + l15)        * DMODEL + khalf;
    const float* pb3 = W + (size_t)(48 + l15)        * DMODEL + khalf;

    const v8f vzero = {0.f, 0.f, 0.f, 0.f, 0.f, 0.f, 0.f, 0.f};
    v8f acc00 = vzero, acc01 = vzero, acc02 = vzero, acc03 = vzero;
    v8f acc10 = vzero, acc11 = vzero, acc12 = vzero, acc13 = vzero;

    // Phase 1: with h-stream prefetch (branch-free hot loop)
    #pragma unroll 4
    for (int k0 = 0; k0 < DMODEL - PFDIST; k0 += 4) {
        __builtin_prefetch(pa0 + k0 + PFDIST, 0, 1);
        __builtin_prefetch(pa1 + k0 + PFDIST, 0, 1);

        v2f a0 = *(const v2f*)(pa0 + k0);
        v2f a1 = *(const v2f*)(pa1 + k0);
        v2f b0 = *(const v2f*)(pb0 + k0);
        v2f b1 = *(const v2f*)(pb1 + k0);
        v2f b2 = *(const v2f*)(pb2 + k0);
        v2f b3 = *(const v2f*)(pb3 + k0);

        acc00 = __builtin_amdgcn_wmma_f32_16x16x4_f32(false, a0, false, b0, (short)0, acc00, false, false);
        acc01 = __builtin_amdgcn_wmma_f32_16x16x4_f32(false, a0, false, b1, (short)0, acc01, false, false);
        acc02 = __builtin_amdgcn_wmma_f32_16x16x4_f32(false, a0, false, b2, (short)0, acc02, false, false);
        acc03 = __builtin_amdgcn_wmma_f32_16x16x4_f32(false, a0, false, b3, (short)0, acc03, false, false);
        acc10 = __builtin_amdgcn_wmma_f32_16x16x4_f32(false, a1, false, b0, (short)0, acc10, false, false);
        acc11 = __builtin_amdgcn_wmma_f32_16x16x4_f32(false, a1, false, b1, (short)0, acc11, false, false);
        acc12 = __builtin_amdgcn_wmma_f32_16x16x4_f32(false, a1, false, b2, (short)0, acc12, false, false);
        acc13 = __builtin_amdgcn_wmma_f32_16x16x4_f32(false, a1, false, b3, (short)0, acc13, false, false);
    }

    // Phase 2: tail without prefetch (stays within the row)
    #pragma unroll 4
    for (int k0 = DMODEL - PFDIST; k0 < DMODEL; k0 += 4) {
        v2f a0 = *(const v2f*)(pa0 + k0);
        v2f a1 = *(const v2f*)(pa1 + k0);
        v2f b0 = *(const v2f*)(pb0 + k0);
        v2f b1 = *(const v2f*)(pb1 + k0);
        v2f b2 = *(const v2f*)(pb2 + k0);
        v2f b3 = *(const v2f*)(pb3 + k0);

        acc00 = __builtin_amdgcn_wmma_f32_16x16x4_f32(false, a0, false, b0, (short)0, acc00, false, false);
        acc01 = __builtin_amdgcn_wmma_f32_16x16x4_f32(false, a0, false, b1, (short)0, acc01, false, false);
        acc02 = __builtin_amdgcn_wmma_f32_16x16x4_f32(false, a0, false, b2, (short)0, acc02, false, false);
        acc03 = __builtin_amdgcn_wmma_f32_16x16x4_f32(false, a0, false, b3, (short)0, acc03, false, false);
        acc10 = __builtin_amdgcn_wmma_f32_16x16x4_f32(false, a1, false, b0, (short)0, acc10, false, false);
        acc11 = __builtin_amdgcn_wmma_f32_16x16x4_f32(false, a1, false, b1, (short)0, acc11, false, false);
        acc12 = __builtin_amdgcn_wmma_f32_16x16x4_f32(false, a1, false, b2, (short)0, acc12, false, false);
        acc13 = __builtin_amdgcn_wmma_f32_16x16x4_f32(false, a1, false, b3, (short)0, acc13, false, false);
    }

    // store D tiles: per VGPR g, 16 lanes write 16 contiguous floats of a row
    const int rofs = (lane >> 4) * 8;            // 0 or 8
    float* out = logits_clean + (size_t)row0 * NEXP;
    #pragma unroll
    for (int g = 0; g < 8; ++g) {
        const size_t r0 = (size_t)(g + rofs) * NEXP;
        out[r0 +      l15] = acc00[g];
        out[r0 + 16 + l15] = acc01[g];
        out[r0 + 32 + l15] = acc02[g];
        out[r0 + 48 + l15] = acc03[g];
        const size_t r1 = (size_t)(16 + g + rofs) * NEXP;
        out[r1 +      l15] = acc10[g];
        out[r1 + 16 + l15] = acc11[g];
        out[r1 + 32 + l15] = acc12[g];
        out[r1 + 48 + l15] = acc13[g];
    }
}

// ---------------------------------------------------------------------------
// Kernel 2: per-row gumbel noise, top-8 mask, masked renormalized softmax.
// One wave32 per row; each lane owns experts 2*lane and 2*lane+1.
// ---------------------------------------------------------------------------
__global__ __launch_bounds__(256) void router_topk_softmax_kernel(
    const float* __restrict__ logits_clean,
    const float* __restrict__ u,
    float* __restrict__ mask_out,
    float* __restrict__ probs_out,
    float* __restrict__ sel_out)
{
    const int lane = threadIdx.x & 31;
    const int wave = threadIdx.x >> 5;
    const int row  = blockIdx.x * 8 + wave;
    const size_t base = (size_t)row * NEXP + 2 * lane;

    v2f lc = *(const v2f*)(logits_clean + base);
    v2f uu = *(const v2f*)(u + base);

    // gumbel: g = -log(-log(clip(u, eps, 1-eps)));  tau = 1, t_sel = 1
    const float eps = 1e-6f;
    float u0 = fminf(fmaxf(uu.x, eps), 1.f - eps);
    float u1 = fminf(fmaxf(uu.y, eps), 1.f - eps);
    float s0 = lc.x - __logf(-__logf(u0));
    float s1 = lc.y - __logf(-__logf(u1));

    v2f sel = {s0, s1};
    *(v2f*)(sel_out + base) = sel;

    // top-8 of 64: 8 rounds of wave-wide argmax (ties -> lowest index)
    float w0 = s0, w1 = s1;
    float m0 = 0.f, m1 = 0.f;
    for (int it = 0; it < TOPK; ++it) {
        float v;
        int idx;
        if (w0 >= w1) { v = w0; idx = 2 * lane; }
        else          { v = w1; idx = 2 * lane + 1; }
        #pragma unroll
        for (int o = 16; o > 0; o >>= 1) {
            float ov = __shfl_xor(v, o, 32);
            int   oi = __shfl_xor(idx, o, 32);
            if (ov > v || (ov == v && oi < idx)) { v = ov; idx = oi; }
        }
        if (idx == 2 * lane)          { m0 = 1.f; w0 = -__builtin_inff(); }
        else if (idx == 2 * lane + 1) { m1 = 1.f; w1 = -__builtin_inff(); }
    }

    // stable softmax over clean logits (t_mix = 1)
    float rmax = fmaxf(lc.x, lc.y);
    #pragma unroll
    for (int o = 16; o > 0; o >>= 1) rmax = fmaxf(rmax, __shfl_xor(rmax, o, 32));
    float e0 = __expf(lc.x - rmax);
    float e1 = __expf(lc.y - rmax);
    float esum = e0 + e1;
    #pragma unroll
    for (int o = 16; o > 0; o >>= 1) esum += __shfl_xor(esum, o, 32);
    float d0 = e0 / esum;
    float d1 = e1 / esum;

    // mask + renormalize
    float p0 = m0 * d0, p1 = m1 * d1;
    float psum = p0 + p1;
    #pragma unroll
    for (int o = 16; o > 0; o >>= 1) psum += __shfl_xor(psum, o, 32);
    float inv = 1.f / (psum + 1e-9f);

    v2f pm = {m0, m1};
    v2f pp = {p0 * inv, p1 * inv};
    *(v2f*)(mask_out + base)  = pm;
    *(v2f*)(probs_out + base) = pp;
}

extern "C" void kernel_launch(void* const* d_in, const int* in_sizes, int n_in,
                              void* d_out, int out_size, void* d_ws, size_t ws_size,
                              hipStream_t stream) {
    const float* h = (const float*)d_in[0];   // (T, D)
    const float* W = (const float*)d_in[1];   // (E, D)
    const float* u = (const float*)d_in[2];   // (T, E)

    float* out = (float*)d_out;
    const size_t TE = (size_t)T_TOK * NEXP;
    float* mask_out  = out;            // (T, E) as 0.0/1.0
    float* probs_out = out + TE;       // (T, E)
    float* clean_out = out + 2 * TE;   // (T, E)
    float* sel_out   = out + 3 * TE;   // (T, E)

    // 256 rows per block (8 waves x 32 rows), all 64 experts per wave
    router_gemm_wmma_kernel<<<dim3(T_TOK / 256), dim3(256), 0, stream>>>(h, W, clean_out);
    // 8 rows per block (one wave32 per row)
    router_topk_softmax_kernel<<<dim3(T_TOK / 8), dim3(256), 0, stream>>>(
        clean_out, u, mask_out, probs_out, sel_out);
}